// HarrisNet_13958643712039
// MI455X (gfx1250) — compile-verified
//
#include <hip/hip_runtime.h>
#include <cstdint>

#define IMG 4096
#define ALPHA 0.05f

typedef __attribute__((ext_vector_type(2))) float v2f;
typedef __attribute__((ext_vector_type(8))) float v8f;

// ---------------- Kernel 1: fused Sobel + products + separable 7x7 Gaussian + Harris R
// block = 256 threads = 8 waves; output tile 64 rows x 32 cols (8 x 16x16 WMMA subtiles)
#define T_ROWS 64
#define T_COLS 32
#define XS_H 72
#define XS_W 40
#define P_H 70
#define P_W 38
#define HB_H 70
#define HB_W 32

__global__ __launch_bounds__(256) void k_harris(const float* __restrict__ x,
                                                const float* __restrict__ G,
                                                float* __restrict__ R)
{
    // smemA holds xs (72x40) first, then is reused for hb (3 x 70x32)
    __shared__ __attribute__((aligned(16))) float smemA[3 * HB_H * HB_W]; // 6720 floats
    __shared__ float smemB[3 * P_H * P_W];                                // 7980 floats
    __shared__ float s_h[8];
    __shared__ float s_acoef[40];   // zero-padded 1-D gaussian for WMMA A matrix

    float* xs = smemA;
    float* hb = smemA;
    float* p  = smemB;

    const int tid  = threadIdx.x;
    const int row0 = blockIdx.y * T_ROWS;
    const int col0 = blockIdx.x * T_COLS;

    // 1-D gaussian factor: column sums of normalized 7x7 kernel (rows sum to 1)
    if (tid < 7) {
        float s = 0.f;
        for (int r = 0; r < 7; ++r) s += G[r * 7 + tid];
        s_h[tid] = s;
    }
    if (tid == 7) s_h[7] = 0.f;
    if (tid < 40) {
        int t = tid - 16;
        float s = 0.f;
        if (t >= 0 && t < 7) {
            for (int r = 0; r < 7; ++r) s += G[r * 7 + t];
        }
        s_acoef[tid] = s;
    }

    // ---- stage x tile (zero-padded outside image)
    const bool interior = (row0 >= 4) && (row0 + XS_H - 4 <= IMG) &&
                          (col0 >= 4) && (col0 + XS_W - 4 <= IMG);
    if (interior) {
        // direct memory->LDS via async DMA engine (B128 per lane), GVS addressing
        const float* base = x + (size_t)(row0 - 4) * IMG + (col0 - 4);
        unsigned long long sbase = (unsigned long long)(uintptr_t)base;
        uint32_t lds0 = (uint32_t)(uintptr_t)xs;
        for (int i4 = tid; i4 < (XS_H * XS_W) / 4; i4 += 256) {
            int r = i4 / (XS_W / 4), c4 = i4 % (XS_W / 4);
            uint32_t voff = (uint32_t)(((size_t)r * IMG + (size_t)c4 * 4) * 4);
            uint32_t vlds = lds0 + (uint32_t)i4 * 16u;
            asm volatile("global_load_async_to_lds_b128 %0, %1, %2"
                         :: "v"(vlds), "v"(voff), "s"(sbase) : "memory");
        }
        asm volatile("s_wait_asynccnt 0x0" ::: "memory");
    } else {
        for (int i = tid; i < XS_H * XS_W; i += 256) {
            int r = i / XS_W, c = i % XS_W;
            int gr = row0 - 4 + r, gc = col0 - 4 + c;
            float v = 0.f;
            if (gr >= 0 && gr < IMG && gc >= 0 && gc < IMG) v = x[gr * IMG + gc];
            xs[i] = v;
        }
    }
    __syncthreads();

    // ---- Sobel (cross-correlation) + products; zero outside image (conv zero-pads prods)
    for (int i = tid; i < P_H * P_W; i += 256) {
        int r = i / P_W, c = i % P_W;
        int gr = row0 - 3 + r, gc = col0 - 3 + c;
        float ix = 0.f, iy = 0.f;
        if (gr >= 0 && gr < IMG && gc >= 0 && gc < IMG) {
            const float* q = xs + (r + 1) * XS_W + (c + 1);
            float a  = q[-XS_W - 1], b  = q[-XS_W], cc = q[-XS_W + 1];
            float d  = q[-1],                        f  = q[+1];
            float g2 = q[XS_W - 1],  hh = q[XS_W],   ii = q[XS_W + 1];
            ix = (cc - a) + 2.f * (f - d) + (ii - g2);
            iy = (g2 - a) + 2.f * (hh - b) + (ii - cc);
        }
        p[0 * P_H * P_W + i] = ix * ix;
        p[1 * P_H * P_W + i] = iy * iy;
        p[2 * P_H * P_W + i] = ix * iy;
    }
    __syncthreads();   // xs dead from here; smemA reused as hb

    // ---- horizontal blur (VALU): hb[ch][r][c] = sum_t p[ch][r][c+t]*h[t]
    float h0 = s_h[0], h1 = s_h[1], h2 = s_h[2], h3 = s_h[3],
          h4 = s_h[4], h5 = s_h[5], h6 = s_h[6];
    for (int i = tid; i < HB_H * HB_W; i += 256) {
        int r = i / HB_W, c = i % HB_W;
        for (int ch = 0; ch < 3; ++ch) {
            const float* pr = p + ch * P_H * P_W + r * P_W + c;
            float s = pr[0] * h0 + pr[1] * h1 + pr[2] * h2 + pr[3] * h3 +
                      pr[4] * h4 + pr[5] * h5 + pr[6] * h6;
            hb[ch * HB_H * HB_W + i] = s;
        }
    }
    __syncthreads();

    // ---- vertical blur as 16x16 matmul: Out = A(16x24 banded coeffs) * B(24x16 hb columns)
    const int wave  = tid >> 5;
    const int lane  = tid & 31;
    const int tr    = wave >> 1;          // 0..3
    const int tc    = wave & 1;           // 0..1
    const int mn    = lane & 15;          // m for A, n for B
    const int khalf = lane >> 4;

    v8f acc0 = {0.f,0.f,0.f,0.f,0.f,0.f,0.f,0.f};
    v8f acc1 = {0.f,0.f,0.f,0.f,0.f,0.f,0.f,0.f};
    v8f acc2 = {0.f,0.f,0.f,0.f,0.f,0.f,0.f,0.f};

#if defined(__has_builtin) && __has_builtin(__builtin_amdgcn_wmma_f32_16x16x4_f32)
    for (int kk = 0; kk < 6; ++kk) {
        int k0 = kk * 4 + khalf * 2;
        // A[m][j] = h[j-m] for 0<=j-m<=6 else 0, via padded table (no branches)
        float ax = s_acoef[16 + k0 - mn];
        float ay = s_acoef[17 + k0 - mn];
        v2f a = {ax, ay};
        // B[j][n] = hb[tr*16 + j][tc*16 + n]; clamp padded rows (coeff is 0 there)
        int rb0 = tr * 16 + k0;     if (rb0 > HB_H - 1) rb0 = HB_H - 1;
        int rb1 = tr * 16 + k0 + 1; if (rb1 > HB_H - 1) rb1 = HB_H - 1;
        int c0i = rb0 * HB_W + tc * 16 + mn;
        int c1i = rb1 * HB_W + tc * 16 + mn;
        v2f b0 = {hb[0 * HB_H * HB_W + c0i], hb[0 * HB_H * HB_W + c1i]};
        v2f b1 = {hb[1 * HB_H * HB_W + c0i], hb[1 * HB_H * HB_W + c1i]};
        v2f b2 = {hb[2 * HB_H * HB_W + c0i], hb[2 * HB_H * HB_W + c1i]};
        acc0 = __builtin_amdgcn_wmma_f32_16x16x4_f32(false, a, false, b0, (short)0, acc0, false, false);
        acc1 = __builtin_amdgcn_wmma_f32_16x16x4_f32(false, a, false, b1, (short)0, acc1, false, false);
        acc2 = __builtin_amdgcn_wmma_f32_16x16x4_f32(false, a, false, b2, (short)0, acc2, false, false);
    }
#else
    // VALU fallback producing the same C/D lane layout
    for (int i = 0; i < 8; ++i) {
        int rloc = tr * 16 + i + 8 * khalf;
        float s0 = 0.f, s1 = 0.f, s2 = 0.f;
        for (int t = 0; t < 7; ++t) {
            float w = s_h[t];
            int idx = (rloc + t) * HB_W + tc * 16 + mn;
            s0 += hb[0 * HB_H * HB_W + idx] * w;
            s1 += hb[1 * HB_H * HB_W + idx] * w;
            s2 += hb[2 * HB_H * HB_W + idx] * w;
        }
        acc0[i] = s0; acc1[i] = s1; acc2[i] = s2;
    }
#endif

    // ---- Harris response from D layout: VGPR i -> row i + 8*khalf, col = n
    for (int i = 0; i < 8; ++i) {
        float sx2 = acc0[i], sy2 = acc1[i], sxy = acc2[i];
        float trc = sx2 + sy2;
        float rv  = sx2 * sy2 - sxy * sxy - ALPHA * trc * trc;
        int gr = row0 + tr * 16 + i + 8 * khalf;
        int gc = col0 + tc * 16 + mn;
        R[gr * IMG + gc] = rv;
    }
}

// ---------------- Median via 3-pass radix select on order-preserved bits (11+11+10)
__global__ void k_init(uint32_t* bins, uint32_t* state)
{
    for (int i = threadIdx.x; i < 3 * 2048; i += 256) bins[i] = 0u;
    if (threadIdx.x < 8) state[threadIdx.x] = 0u;
}

__global__ __launch_bounds__(256) void k_hist(const float* __restrict__ R,
                                              uint32_t* __restrict__ bins,
                                              const uint32_t* __restrict__ state,
                                              int pass)
{
    __shared__ uint32_t lb[2048];
    for (int i = threadIdx.x; i < 2048; i += 256) lb[i] = 0u;
    __syncthreads();
    uint32_t prefix = state[0];
    const long long N = (long long)IMG * IMG;
    long long stride = (long long)gridDim.x * blockDim.x;
    for (long long i = (long long)blockIdx.x * blockDim.x + threadIdx.x; i < N; i += stride) {
        uint32_t s = __float_as_uint(R[i]);
        uint32_t u = (s & 0x80000000u) ? ~s : (s | 0x80000000u);
        if (pass == 0) {
            atomicAdd(&lb[u >> 21], 1u);
        } else if (pass == 1) {
            if ((u >> 21) == prefix) atomicAdd(&lb[(u >> 10) & 2047u], 1u);
        } else {
            if ((u >> 10) == prefix) atomicAdd(&lb[u & 1023u], 1u);
        }
    }
    __syncthreads();
    for (int i = threadIdx.x; i < 2048; i += 256) {
        uint32_t c = lb[i];
        if (c) atomicAdd(&bins[i], c);
    }
}

__global__ void k_select(const uint32_t* bins, uint32_t* state, int pass)
{
    if (threadIdx.x != 0 || blockIdx.x != 0) return;
    unsigned long long k = (pass == 0) ? (((unsigned long long)IMG * IMG - 1ull) / 2ull)
                                       : (unsigned long long)state[1];
    int nb = (pass == 2) ? 1024 : 2048;
    unsigned long long cum = 0; uint32_t sel = 0; bool found = false;
    for (int b = 0; b < nb; ++b) {
        unsigned long long c = bins[b];
        if (!found && cum + c > k) { sel = (uint32_t)b; found = true; }
        if (!found) cum += c;
    }
    k -= cum;
    if (pass == 0) { state[0] = sel; state[1] = (uint32_t)k; }
    else if (pass == 1) { state[0] = (state[0] << 11) | sel; state[1] = (uint32_t)k; }
    else {
        uint32_t u = (state[0] << 10) | sel;
        state[2] = u;
        state[3] = (u & 0x80000000u) ? (u & 0x7fffffffu) : ~u;   // float bits of median
    }
}

// ---------------- NMS: threshold + separable 7x7 max, emit (Rt==Rmax) as wave32 ballot bits
#define NB_W 32
#define NB_H 8
__global__ __launch_bounds__(256) void k_nms(const float* __restrict__ R,
                                             const uint32_t* __restrict__ state,
                                             uint32_t* __restrict__ maskw)
{
    __shared__ float rt[(NB_H + 6) * (NB_W + 6)];  // 14 x 38
    __shared__ float hm[(NB_H + 6) * NB_W];        // 14 x 32
    const float med = __uint_as_float(state[3]);
    const int gx0 = blockIdx.x * NB_W, gy0 = blockIdx.y * NB_H;
    const int tid = threadIdx.y * NB_W + threadIdx.x;

    for (int i = tid; i < 14 * 38; i += 256) {
        int r = i / 38, c = i % 38;
        int gy = gy0 - 3 + r, gx = gx0 - 3 + c;
        float v = -__builtin_inff();
        if (gy >= 0 && gy < IMG && gx >= 0 && gx < IMG) {
            float rv = R[gy * IMG + gx];
            v = (rv < med) ? 0.f : rv;
        }
        rt[i] = v;
    }
    __syncthreads();
    for (int i = tid; i < 14 * 32; i += 256) {
        int r = i / 32, c = i % 32;
        const float* q = rt + r * 38 + c;
        float mx = q[0];
        for (int t = 1; t < 7; ++t) mx = fmaxf(mx, q[t]);
        hm[r * 32 + c] = mx;
    }
    __syncthreads();
    float rmax = hm[threadIdx.y * 32 + threadIdx.x];
    for (int t = 1; t < 7; ++t) rmax = fmaxf(rmax, hm[(threadIdx.y + t) * 32 + threadIdx.x]);
    float rc = rt[(threadIdx.y + 3) * 38 + threadIdx.x + 3];
    unsigned long long bal = __ballot(rc == rmax);
    if (threadIdx.x == 0) {
        int gy = gy0 + threadIdx.y;
        maskw[gy * (IMG / 32) + blockIdx.x] = (uint32_t)bal;
    }
}

__global__ void k_final(float* __restrict__ R,
                        const uint32_t* __restrict__ maskw,
                        const uint32_t* __restrict__ state)
{
    const float med = __uint_as_float(state[3]);
    long long i = (long long)blockIdx.x * blockDim.x + threadIdx.x;
    float r  = R[i];
    float rt = (r < med) ? 0.f : r;
    uint32_t w = maskw[i >> 5];
    bool bit = (w >> ((int)i & 31)) & 1u;
    R[i] = bit ? rt : 0.f;
}

// ---------------- host launcher
extern "C" void kernel_launch(void* const* d_in, const int* in_sizes, int n_in,
                              void* d_out, int out_size, void* d_ws, size_t ws_size,
                              hipStream_t stream)
{
    (void)in_sizes; (void)n_in; (void)out_size; (void)ws_size;
    const float* x = (const float*)d_in[0];
    const float* G = (const float*)d_in[1];
    float* R = (float*)d_out;                      // scratch for R, then final output in place

    uint32_t* bins  = (uint32_t*)d_ws;             // 3 * 2048 bins
    uint32_t* state = bins + 3 * 2048;             // 8 words
    uint32_t* maskw = (uint32_t*)((char*)d_ws + 32768);   // 4096*4096/32 words = 2MB

    k_init<<<1, 256, 0, stream>>>(bins, state);

    dim3 hg(IMG / T_COLS, IMG / T_ROWS);           // (128, 64)
    k_harris<<<hg, 256, 0, stream>>>(x, G, R);

    for (int pass = 0; pass < 3; ++pass) {
        k_hist<<<2048, 256, 0, stream>>>(R, bins + pass * 2048, state, pass);
        k_select<<<1, 1, 0, stream>>>(bins + pass * 2048, state, pass);
    }

    dim3 nb(NB_W, NB_H), ng(IMG / NB_W, IMG / NB_H);
    k_nms<<<ng, nb, 0, stream>>>(R, state, maskw);

    k_final<<<(IMG * (long long)IMG) / 256, 256, 0, stream>>>(R, maskw, state);
}